// HeteroGCNWithEdgeWeights_84988812853628
// MI455X (gfx1250) — compile-verified
//
#include <hip/hip_runtime.h>

typedef float v2f __attribute__((ext_vector_type(2)));
typedef float v8f __attribute__((ext_vector_type(8)));

static constexpr int kNodes = 100000;
static constexpr int kEdges = 1600000;
static constexpr int kIn    = 128;
static constexpr int kHid   = 128;
static constexpr int kCls   = 2;
static constexpr int kRel   = 3;

__device__ __forceinline__ void atomAddF(float* p, float v) {
  // lowers to global_atomic_add_f32 (no CAS loop)
  unsafeAtomicAdd(p, v);
}

__global__ void zero_kernel(float* __restrict__ p, int n) {
  int i = blockIdx.x * blockDim.x + threadIdx.x;
  int stride = gridDim.x * blockDim.x;
  for (; i < n; i += stride) p[i] = 0.0f;
}

// deg[r*N + dst] += 1 over all (r, e)
__global__ void degree_kernel(const int* __restrict__ dst, float* __restrict__ deg) {
  int i = blockIdx.x * blockDim.x + threadIdx.x;
  if (i >= kRel * kEdges) return;
  int r = i / kEdges;
  atomAddF(&deg[(size_t)r * kNodes + dst[i]], 1.0f);
}

// deg -> 1 / max(deg, 1)   (DGL norm='right' with clamp(min=1))
__global__ void deginv_kernel(float* __restrict__ deg) {
  int i = blockIdx.x * blockDim.x + threadIdx.x;
  if (i >= kRel * kNodes) return;
  deg[i] = 1.0f / fmaxf(deg[i], 1.0f);
}

// h = x @ W ; x:[kNodes,128] row-major, W:[128,128] row-major (staged in LDS).
// One wave computes a 16-row x 128-col stripe via V_WMMA_F32_16X16X4_F32.
// A frag (16x4 f32): lanes 0-15 hold K={k,k+1} of row M=lane; lanes 16-31 hold K={k+2,k+3}.
// B frag (4x16 f32): VGPR0 = row k (lanes 0-15) / row k+2 (lanes 16-31); VGPR1 = rows k+1 / k+3.
// C/D (16x16 f32):   VGPR v = row v (lanes 0-15) / row v+8 (lanes 16-31), col = lane&15.
__global__ void __launch_bounds__(256) gemm_node_kernel(const float* __restrict__ x,
                                                        const float* __restrict__ W,
                                                        float* __restrict__ h) {
  __shared__ float sW[kIn * kHid];  // 64 KB of the 320 KB WGP LDS
  {
    const float4* s4 = (const float4*)W;
    float4*       d4 = (float4*)sW;
    for (int i = threadIdx.x; i < (kIn * kHid) / 4; i += 256) d4[i] = s4[i];
  }
  __syncthreads();

  const int wave = threadIdx.x >> 5;
  const int lane = threadIdx.x & 31;
  const int half = lane >> 4;
  const int mrow = lane & 15;
  const int tile = blockIdx.x * 8 + wave;   // 16-node tile; 100000/16 = 6250 exactly
  if (tile * 16 >= kNodes) return;
  const int m0 = tile * 16;

  v8f acc[8] = {};
  const float* arow = x + (size_t)(m0 + mrow) * kIn;

#pragma unroll 2
  for (int k = 0; k < kIn; k += 4) {
    const int kk = k + half * 2;
    v2f a = *(const v2f*)(arow + kk);
#pragma unroll
    for (int nt = 0; nt < 8; ++nt) {
      v2f b;
      b.x = sW[kk * kHid + nt * 16 + mrow];
      b.y = sW[(kk + 1) * kHid + nt * 16 + mrow];
      acc[nt] = __builtin_amdgcn_wmma_f32_16x16x4_f32(
          /*neg_a=*/false, a, /*neg_b=*/false, b,
          /*c_mod=*/(short)0, acc[nt], /*reuse_a=*/false, /*reuse_b=*/false);
    }
  }

#pragma unroll
  for (int nt = 0; nt < 8; ++nt) {
#pragma unroll
    for (int v = 0; v < 8; ++v) {
      h[(size_t)(m0 + v + half * 8) * kHid + nt * 16 + mrow] = acc[nt][v];
    }
  }
}

// One wave per edge: out[dst] += h[src] * w * deg_inv[dst]  (128 channels, 4/lane)
__global__ void scatter1_kernel(const float* __restrict__ h, const float* __restrict__ w,
                                const int* __restrict__ src, const int* __restrict__ dst,
                                const float* __restrict__ dinv, float* __restrict__ out) {
  int wid  = (blockIdx.x * blockDim.x + threadIdx.x) >> 5;
  int lane = threadIdx.x & 31;
  if (wid >= kEdges) return;
  int s = src[wid];
  int d = dst[wid];
  float scale = w[wid] * dinv[d];
  float4 v = ((const float4*)(h + (size_t)s * kHid))[lane];
  float* op = out + (size_t)d * kHid + lane * 4;
  atomAddF(op + 0, v.x * scale);
  atomAddF(op + 1, v.y * scale);
  atomAddF(op + 2, v.z * scale);
  atomAddF(op + 3, v.w * scale);
}

// out1 = relu(out1 + sum_r b1[r])
__global__ void bias_relu_kernel(float* __restrict__ out1, const float* __restrict__ b1) {
  int i = blockIdx.x * blockDim.x + threadIdx.x;
  if (i >= kNodes * kHid) return;
  int c = i & (kHid - 1);
  float v = out1[i] + b1[c] + b1[kHid + c] + b1[2 * kHid + c];
  out1[i] = v > 0.0f ? v : 0.0f;
}

// h2[n, 0:2] = hin[n, :] @ W2r   (K=128, N=2 -> plain VALU dot, W2 in LDS)
__global__ void __launch_bounds__(256) gemm_cls_kernel(const float* __restrict__ hin,
                                                       const float* __restrict__ W2r,
                                                       float* __restrict__ h2) {
  __shared__ float sw[kHid * kCls];  // 256 floats == blockDim
  sw[threadIdx.x] = W2r[threadIdx.x];
  __syncthreads();
  int n = blockIdx.x * blockDim.x + threadIdx.x;
  if (n >= kNodes) return;
  const float4* hp = (const float4*)(hin + (size_t)n * kHid);
  float a0 = 0.0f, a1 = 0.0f;
#pragma unroll 4
  for (int k4 = 0; k4 < kHid / 4; ++k4) {
    float4 v = hp[k4];
    int k = k4 * 4;
    a0 += v.x * sw[(k + 0) * 2] + v.y * sw[(k + 1) * 2] +
          v.z * sw[(k + 2) * 2] + v.w * sw[(k + 3) * 2];
    a1 += v.x * sw[(k + 0) * 2 + 1] + v.y * sw[(k + 1) * 2 + 1] +
          v.z * sw[(k + 2) * 2 + 1] + v.w * sw[(k + 3) * 2 + 1];
  }
  h2[(size_t)n * kCls + 0] = a0;
  h2[(size_t)n * kCls + 1] = a1;
}

// One thread per edge: out[dst, 0:2] += h2[src, 0:2] * w * deg_inv[dst]
__global__ void scatter2_kernel(const float* __restrict__ h2, const float* __restrict__ w,
                                const int* __restrict__ src, const int* __restrict__ dst,
                                const float* __restrict__ dinv, float* __restrict__ out) {
  int e = blockIdx.x * blockDim.x + threadIdx.x;
  if (e >= kEdges) return;
  int s = src[e];
  int d = dst[e];
  float scale = w[e] * dinv[d];
  atomAddF(out + (size_t)d * kCls + 0, h2[(size_t)s * kCls + 0] * scale);
  atomAddF(out + (size_t)d * kCls + 1, h2[(size_t)s * kCls + 1] * scale);
}

// out += sum_r b2[r]
__global__ void bias2_kernel(float* __restrict__ out, const float* __restrict__ b2) {
  int i = blockIdx.x * blockDim.x + threadIdx.x;
  if (i >= kNodes * kCls) return;
  int c = i & (kCls - 1);
  out[i] += b2[c] + b2[kCls + c] + b2[2 * kCls + c];
}

extern "C" void kernel_launch(void* const* d_in, const int* in_sizes, int n_in,
                              void* d_out, int out_size, void* d_ws, size_t ws_size,
                              hipStream_t stream) {
  (void)in_sizes; (void)n_in; (void)out_size; (void)ws_size;
  const float* x   = (const float*)d_in[0];
  const float* w   = (const float*)d_in[1];
  const float* W1  = (const float*)d_in[2];
  const float* b1  = (const float*)d_in[3];
  const float* W2  = (const float*)d_in[4];
  const float* b2  = (const float*)d_in[5];
  const int*   src = (const int*)d_in[6];
  const int*   dst = (const int*)d_in[7];
  float* out = (float*)d_out;

  // Workspace layout (floats); all region starts are multiples of 4 floats (16 B).
  float* ws   = (float*)d_ws;
  float* deg  = ws;                                   // kRel * kNodes
  float* out1 = deg + (size_t)kRel * kNodes;          // kNodes * kHid
  float* htmp = out1 + (size_t)kNodes * kHid;         // kNodes * kHid
  float* h2   = htmp + (size_t)kNodes * kHid;         // kNodes * kCls

  const int T = 256;
  auto blocks = [](long n, int t) { return (int)((n + t - 1) / t); };

  // 0) zero accumulators (required every call: harness does not re-poison)
  zero_kernel<<<blocks((long)kRel * kNodes, T), T, 0, stream>>>(deg, kRel * kNodes);
  zero_kernel<<<blocks((long)kNodes * kHid, T), T, 0, stream>>>(out1, kNodes * kHid);
  zero_kernel<<<blocks((long)kNodes * kCls, T), T, 0, stream>>>(out, kNodes * kCls);

  // 1) degrees -> reciprocal clamped degrees
  degree_kernel<<<blocks((long)kRel * kEdges, T), T, 0, stream>>>(dst, deg);
  deginv_kernel<<<blocks((long)kRel * kNodes, T), T, 0, stream>>>(deg);

  // 2) layer 1: per relation, WMMA GEMM then edge scatter into out1
  const int gemmBlocks = (kNodes / 16 + 7) / 8;  // 6250 wave-tiles, 8 waves/block
  for (int r = 0; r < kRel; ++r) {
    gemm_node_kernel<<<gemmBlocks, T, 0, stream>>>(
        x, W1 + (size_t)r * kIn * kHid, htmp);
    scatter1_kernel<<<blocks((long)kEdges * 32, T), T, 0, stream>>>(
        htmp, w + (size_t)r * kEdges, src + (size_t)r * kEdges,
        dst + (size_t)r * kEdges, deg + (size_t)r * kNodes, out1);
  }

  // 3) bias + relu (in place)
  bias_relu_kernel<<<blocks((long)kNodes * kHid, T), T, 0, stream>>>(out1, b1);

  // 4) layer 2: per relation, 128->2 GEMM then edge scatter into d_out
  for (int r = 0; r < kRel; ++r) {
    gemm_cls_kernel<<<blocks(kNodes, T), T, 0, stream>>>(
        out1, W2 + (size_t)r * kHid * kCls, h2);
    scatter2_kernel<<<blocks(kEdges, T), T, 0, stream>>>(
        h2, w + (size_t)r * kEdges, src + (size_t)r * kEdges,
        dst + (size_t)r * kEdges, deg + (size_t)r * kNodes, out);
  }

  // 5) final bias
  bias2_kernel<<<blocks((long)kNodes * kCls, T), T, 0, stream>>>(out, b2);
}